// GNNClassifier_88648124990426
// MI455X (gfx1250) — compile-verified
//
#include <hip/hip_runtime.h>
#include <hip/hip_bf16.h>

// GNN SAGE classifier for gfx1250 (MI455X).
// Heavy cost = edge scatter (L2-resident global_atomic_add_f32);
// node transforms use V_WMMA_F32_16X16X4_F32 (f32 SGEMM matrix op).

typedef __attribute__((ext_vector_type(2))) float v2f;
typedef __attribute__((ext_vector_type(8))) float v8f;

// ---------------------------------------------------------------------------
// h[n, 0:64] = emb[x[n], 0:64]   (float4 vectorized)
__global__ void embed_kernel(const int* __restrict__ x,
                             const float4* __restrict__ emb,
                             float4* __restrict__ h, long n16) {
    long i = (long)blockIdx.x * blockDim.x + threadIdx.x;   // over n_nodes*16
    if (i < n16) {
        long n = i >> 4;
        int  f = (int)(i & 15);
        h[i] = emb[(long)x[n] * 16 + f];
    }
}

// deg[dst[e]] += 1
__global__ void degree_kernel(const int* __restrict__ dst,
                              float* __restrict__ deg, long n_edges) {
    long e = (long)blockIdx.x * blockDim.x + threadIdx.x;
    if (e < n_edges) atomicAdd(&deg[dst[e]], 1.0f);
}

// acc[dst[e], f] += h[src[e], f]  — 64 consecutive lanes cover one edge row
__global__ void scatter_add_kernel(const float* __restrict__ h,
                                   const int* __restrict__ src,
                                   const int* __restrict__ dst,
                                   float* __restrict__ acc, long total) {
    long i = (long)blockIdx.x * blockDim.x + threadIdx.x;   // over n_edges*64
    if (i < total) {
        long e = i >> 6;
        int  f = (int)(i & 63);
        int  s = src[e];
        int  d = dst[e];
        atomicAdd(&acc[(long)d * 64 + f], h[(long)s * 64 + f]);
    }
}

// ---------------------------------------------------------------------------
// out[n,:] = relu( (agg[n,:]/max(deg,1)) @ Wl^T + b + h[n,:] @ Wr^T )
// One wave (32 threads) per 16-node tile; K outermost so each A fragment is
// loaded/scaled once and reused by 4 column-tile WMMAs (4 independent
// accumulation chains). `out` may alias `agg`: all global reads for this
// block's 16 rows complete before any store.
//
// WMMA f32 16x16x4 fragment layouts (ISA 7.12.2):
//   A: lane m=lane&15, khalf=2*(lane>>4); A VGPR0=K[khalf], VGPR1=K[khalf+1]
//   B: lane n=lane&15, same khalf;        B VGPR0=B[khalf][n], VGPR1=B[khalf+1][n]
//   C/D: VGPR r -> row (r + 8*(lane>>4)), col = lane&15
__global__ __launch_bounds__(32)
void sage_transform_kernel(const float* __restrict__ h,
                           const float* __restrict__ agg,
                           const float* __restrict__ deg,
                           const float* __restrict__ Wl,   // [64,64] row-major [out,in]
                           const float* __restrict__ bias, // [64]
                           const float* __restrict__ Wr,   // [64,64]
                           float* __restrict__ out) {
    const int lane  = threadIdx.x;
    const int col   = lane & 15;        // N for B/C/D, M for A fragment
    const int hi    = lane >> 4;        // selects K pair {0,1} vs {2,3}
    const int node0 = blockIdx.x * 16;

    const long arow = (long)(node0 + col);
    const float inv = 1.0f / fmaxf(deg[arow], 1.0f);
    const float* __restrict__ Hr = h   + arow * 64 + 2 * hi;
    const float* __restrict__ Ar = agg + arow * 64 + 2 * hi;
    const float* __restrict__ WlC = Wl + (long)col * 64 + 2 * hi;  // + t*16*64 per tile
    const float* __restrict__ WrC = Wr + (long)col * 64 + 2 * hi;

    v8f acc[4];
#pragma unroll
    for (int t = 0; t < 4; ++t) {
        const float bj = bias[t * 16 + col];
#pragma unroll
        for (int r = 0; r < 8; ++r) acc[t][r] = bj;
    }

#pragma unroll
    for (int k = 0; k < 64; k += 4) {
        v2f am, ah;
        am.x = Ar[k]     * inv;         // mean-aggregated neighbor fragment
        am.y = Ar[k + 1] * inv;
        ah.x = Hr[k];                   // self-feature fragment
        ah.y = Hr[k + 1];
#pragma unroll
        for (int t = 0; t < 4; ++t) {
            const long wofs = (long)t * 16 * 64 + k;
            v2f bl, br;
            bl.x = WlC[wofs];
            bl.y = WlC[wofs + 1];
            br.x = WrC[wofs];
            br.y = WrC[wofs + 1];
            acc[t] = __builtin_amdgcn_wmma_f32_16x16x4_f32(
                         false, am, false, bl, (short)0, acc[t], false, false);
            acc[t] = __builtin_amdgcn_wmma_f32_16x16x4_f32(
                         false, ah, false, br, (short)0, acc[t], false, false);
        }
    }

    // all reads done -> safe to store even when out aliases agg
#pragma unroll
    for (int t = 0; t < 4; ++t) {
#pragma unroll
        for (int r = 0; r < 8; ++r) {
            float v = fmaxf(acc[t][r], 0.0f);
            out[(long)(node0 + r + 8 * hi) * 64 + t * 16 + col] = v;
        }
    }
}

// ---------------------------------------------------------------------------
// pooled[batch[n], f] += h[n, f];  gcnt[batch[n]] += 1 (once per node)
__global__ void pool_kernel(const float* __restrict__ h,
                            const int* __restrict__ batch,
                            float* __restrict__ pooled,
                            float* __restrict__ gcnt, long total) {
    long i = (long)blockIdx.x * blockDim.x + threadIdx.x;   // over n_nodes*64
    if (i < total) {
        long n = i >> 6;
        int  f = (int)(i & 63);
        int  g = batch[n];
        atomicAdd(&pooled[(long)g * 64 + f], h[i]);
        if (f == 0) atomicAdd(&gcnt[g], 1.0f);
    }
}

// out[g,:2] = (pooled[g,:]/max(gcnt,1)) @ Wout^T + bout
__global__ void head_kernel(const float* __restrict__ pooled,
                            const float* __restrict__ gcnt,
                            const float* __restrict__ Wout,  // [2,64]
                            const float* __restrict__ bout,  // [2]
                            float* __restrict__ out, int n_graphs) {
    int g = blockIdx.x * blockDim.x + threadIdx.x;
    if (g < n_graphs) {
        float inv = 1.0f / fmaxf(gcnt[g], 1.0f);
        float a0 = bout[0], a1 = bout[1];
        const float* p = pooled + (long)g * 64;
#pragma unroll
        for (int k = 0; k < 64; ++k) {
            float v = p[k] * inv;
            a0 = fmaf(v, Wout[k], a0);
            a1 = fmaf(v, Wout[64 + k], a1);
        }
        out[g * 2 + 0] = a0;
        out[g * 2 + 1] = a1;
    }
}

// ---------------------------------------------------------------------------
extern "C" void kernel_launch(void* const* d_in, const int* in_sizes, int n_in,
                              void* d_out, int out_size, void* d_ws, size_t ws_size,
                              hipStream_t stream) {
    const int*   x     = (const int*)d_in[0];
    const int*   ei    = (const int*)d_in[1];
    const int*   batch = (const int*)d_in[2];
    const float* emb   = (const float*)d_in[3];
    const float* W1l   = (const float*)d_in[4];
    const float* b1    = (const float*)d_in[5];
    const float* W1r   = (const float*)d_in[6];
    const float* W2l   = (const float*)d_in[7];
    const float* b2    = (const float*)d_in[8];
    const float* W2r   = (const float*)d_in[9];
    const float* Wout  = (const float*)d_in[10];
    const float* bout  = (const float*)d_in[11];
    float*       out   = (float*)d_out;

    const long n_nodes  = in_sizes[0];
    const long n_edges  = in_sizes[1] / 2;
    const int  n_graphs = out_size / 2;
    const int* src = ei;
    const int* dst = ei + n_edges;

    // workspace layout (floats)
    float* bufA   = (float*)d_ws;                 // [N,64] node features (ping)
    float* bufB   = bufA + n_nodes * 64;          // [N,64] node features (pong)
    float* deg    = bufB + n_nodes * 64;          // [N]
    float* pooled = deg + n_nodes;                // [G,64]
    float* gcnt   = pooled + (long)n_graphs * 64; // [G]

    const int T = 256;
    const long nf   = n_nodes * 64;               // node-feature elements
    const long ef   = n_edges * 64;               // edge-feature work items
    const long n16  = n_nodes * 16;               // float4 elements

    // zero accumulators (memset nodes are graph-capturable)
    hipMemsetAsync(bufB, 0, (size_t)nf * sizeof(float), stream);
    hipMemsetAsync(deg, 0, (size_t)n_nodes * sizeof(float), stream);
    hipMemsetAsync(pooled, 0, (size_t)(n_graphs * 64 + n_graphs) * sizeof(float), stream);

    // h0 = emb[x]
    embed_kernel<<<dim3((n16 + T - 1) / T), dim3(T), 0, stream>>>(
        x, (const float4*)emb, (float4*)bufA, n16);

    // degrees (shared by both convs)
    degree_kernel<<<dim3((n_edges + T - 1) / T), dim3(T), 0, stream>>>(
        dst, deg, n_edges);

    // ---- SAGEConv 1: bufB = relu(mean(scatter(bufA)) @ W1l^T + b1 + bufA @ W1r^T)
    scatter_add_kernel<<<dim3((ef + T - 1) / T), dim3(T), 0, stream>>>(
        bufA, src, dst, bufB, ef);
    sage_transform_kernel<<<dim3(n_nodes / 16), dim3(32), 0, stream>>>(
        bufA, bufB, deg, W1l, b1, W1r, bufB);

    // ---- SAGEConv 2: bufA = relu(mean(scatter(bufB)) @ W2l^T + b2 + bufB @ W2r^T)
    hipMemsetAsync(bufA, 0, (size_t)nf * sizeof(float), stream);
    scatter_add_kernel<<<dim3((ef + T - 1) / T), dim3(T), 0, stream>>>(
        bufB, src, dst, bufA, ef);
    sage_transform_kernel<<<dim3(n_nodes / 16), dim3(32), 0, stream>>>(
        bufB, bufA, deg, W2l, b2, W2r, bufA);

    // ---- global mean pool + linear head
    pool_kernel<<<dim3((nf + T - 1) / T), dim3(T), 0, stream>>>(
        bufA, batch, pooled, gcnt, nf);
    head_kernel<<<dim3((n_graphs + 127) / 128), dim3(128), 0, stream>>>(
        pooled, gcnt, Wout, bout, out, n_graphs);
}